// StandardSlotModel_3204045603462
// MI455X (gfx1250) — compile-verified
//
#include <hip/hip_runtime.h>

#define H        64
#define H2       128
#define VOCAB    64
#define K_SLOTS  8
#define NB       128
#define NL       4096

// LDS row strides (padded to dodge bank conflicts)
#define S0 68    // h0 / h rows: 64 + 4
#define S1 132   // act1 rows: 128 + 4

typedef __attribute__((ext_vector_type(2))) float v2f;
typedef __attribute__((ext_vector_type(8))) float v8f;

#if __has_builtin(__builtin_amdgcn_wmma_f32_16x16x4_f32)
#define HAVE_WMMA_F32 1
#else
#define HAVE_WMMA_F32 0
#endif

// One wave computes h (post-LayerNorm) for a 16-token tile into h0s[16*S0].
// WMMA f32 16x16x4: A lanes0-15 = rows, K pair {0,1}; lanes16-31 K pair {2,3}.
// C layout: VGPR j -> (row j, col lane) for lanes 0-15, (row j+8, col lane-16).
static __device__ __forceinline__ void compute_h_tile(
    const int* tok,
    const float* __restrict__ embed,
    const float* __restrict__ W1, const float* __restrict__ b1,
    const float* __restrict__ W2, const float* __restrict__ b2,
    const float* __restrict__ ln_g, const float* __restrict__ ln_b,
    float* h0s, float* act1, int lane)
{
    const int half = lane >> 4;   // 0 or 1
    const int l16  = lane & 15;

    // ---- gather h0 = embed[tok] into LDS [16][H]
    for (int e = lane; e < 16 * H; e += 32) {
        int r = e >> 6, c = e & 63;
        h0s[r * S0 + c] = embed[tok[r] * H + c];
    }
    __syncthreads();

    // ---- GEMM1: act1[16][128] = relu(h0 @ W1 + b1)
    for (int nt = 0; nt < 8; ++nt) {
        const int col = nt * 16 + l16;
        float bias = b1[col];
        v8f c = {bias, bias, bias, bias, bias, bias, bias, bias};
        #pragma unroll
        for (int k = 0; k < 16; ++k) {
            const int k0 = k * 4 + half * 2;
#if HAVE_WMMA_F32
            v2f a, bf;
            a.x  = h0s[l16 * S0 + k0];
            a.y  = h0s[l16 * S0 + k0 + 1];
            bf.x = W1[(k0)     * H2 + col];
            bf.y = W1[(k0 + 1) * H2 + col];
            c = __builtin_amdgcn_wmma_f32_16x16x4_f32(false, a, false, bf,
                                                      (short)0, c, false, false);
#else
            #pragma unroll
            for (int j = 0; j < 8; ++j) {
                const int row = half * 8 + j;
                #pragma unroll
                for (int kk = 0; kk < 4; ++kk)
                    c[j] += h0s[row * S0 + k * 4 + kk] * W1[(k * 4 + kk) * H2 + col];
            }
#endif
        }
        #pragma unroll
        for (int j = 0; j < 8; ++j) {
            const int row = half * 8 + j;
            float v = c[j];
            act1[row * S1 + col] = v > 0.f ? v : 0.f;
        }
    }
    __syncthreads();

    // ---- GEMM2 + residual: h0s = h0 + act1 @ W2 + b2
    for (int nt = 0; nt < 4; ++nt) {
        const int col = nt * 16 + l16;
        float bias = b2[col];
        v8f c = {bias, bias, bias, bias, bias, bias, bias, bias};
        #pragma unroll
        for (int k = 0; k < 32; ++k) {
            const int k0 = k * 4 + half * 2;
#if HAVE_WMMA_F32
            v2f a, bf;
            a.x  = act1[l16 * S1 + k0];
            a.y  = act1[l16 * S1 + k0 + 1];
            bf.x = W2[(k0)     * H + col];
            bf.y = W2[(k0 + 1) * H + col];
            c = __builtin_amdgcn_wmma_f32_16x16x4_f32(false, a, false, bf,
                                                      (short)0, c, false, false);
#else
            #pragma unroll
            for (int j = 0; j < 8; ++j) {
                const int row = half * 8 + j;
                #pragma unroll
                for (int kk = 0; kk < 4; ++kk)
                    c[j] += act1[row * S1 + k * 4 + kk] * W2[(k * 4 + kk) * H + col];
            }
#endif
        }
        #pragma unroll
        for (int j = 0; j < 8; ++j) {
            const int row = half * 8 + j;
            h0s[row * S0 + col] += c[j];
        }
    }
    __syncthreads();

    // ---- LayerNorm per row (16 rows handled by lanes 0-15)
    if (lane < 16) {
        const float* xr = h0s + lane * S0;
        float mu = 0.f;
        for (int c = 0; c < H; ++c) mu += xr[c];
        mu *= (1.0f / H);
        float var = 0.f;
        for (int c = 0; c < H; ++c) { float d = xr[c] - mu; var += d * d; }
        var *= (1.0f / H);
        float inv = rsqrtf(var + 1e-5f);
        float* wr = h0s + lane * S0;
        for (int c = 0; c < H; ++c)
            wr[c] = (wr[c] - mu) * inv * ln_g[c] + ln_b[c];
    }
    __syncthreads();
}

// ---------------- Pass 1: fused encoder -> gate scores ----------------
__global__ void __launch_bounds__(128)
score_kernel(const int* __restrict__ seq, const float* __restrict__ embed,
             const float* __restrict__ W1, const float* __restrict__ b1,
             const float* __restrict__ W2, const float* __restrict__ b2,
             const float* __restrict__ ln_g, const float* __restrict__ ln_b,
             const float* __restrict__ gate_w, const float* __restrict__ gate_b,
             float* __restrict__ scores)
{
    __shared__ float h0s[4][16 * S0];
    __shared__ float act1[4][16 * S1];
    __shared__ int   tok[4][16];

    const int wave = threadIdx.x >> 5;
    const int lane = threadIdx.x & 31;
    const int base = (blockIdx.x * 4 + wave) * 16;   // flat token index, tile within one row

    if (lane < 16) tok[wave][lane] = seq[base + lane];
    __syncthreads();

    compute_h_tile(tok[wave], embed, W1, b1, W2, b2, ln_g, ln_b,
                   h0s[wave], act1[wave], lane);

    if (lane < 16) {
        float s = gate_b[0];
        const float* hr = h0s[wave] + lane * S0;
        for (int c = 0; c < H; ++c) s += hr[c] * gate_w[c];
        scores[base + lane] = s;
    }
}

// ---------------- Pass 2: per-batch top-8 (iterative argmax) ----------------
__global__ void __launch_bounds__(256)
topk_kernel(float* __restrict__ scores, int* __restrict__ idx_out)
{
    __shared__ float sv[256];
    __shared__ int   si[256];
    const int b = blockIdx.x;
    const int tid = threadIdx.x;
    float* sc = scores + (long)b * NL;

    for (int k = 0; k < K_SLOTS; ++k) {
        float best = -3.402823466e38f; int bi = NL;
        for (int i = tid; i < NL; i += 256) {
            float v = sc[i];
            if (v > best) { best = v; bi = i; }
        }
        sv[tid] = best; si[tid] = bi;
        __syncthreads();
        for (int off = 128; off > 0; off >>= 1) {
            if (tid < off) {
                float v2 = sv[tid + off]; int i2 = si[tid + off];
                if (v2 > sv[tid] || (v2 == sv[tid] && i2 < si[tid])) {
                    sv[tid] = v2; si[tid] = i2;
                }
            }
            __syncthreads();
        }
        if (tid == 0) { idx_out[b * K_SLOTS + k] = si[0]; sc[si[0]] = -3.402823466e38f; }
        __syncthreads();
    }
}

// ---------------- Pass 3: recompute h for 9 tokens, attention head ----------------
__global__ void __launch_bounds__(32)
finalize_kernel(const int* __restrict__ seq, const int* __restrict__ idx,
                const float* __restrict__ embed,
                const float* __restrict__ W1, const float* __restrict__ b1,
                const float* __restrict__ W2, const float* __restrict__ b2,
                const float* __restrict__ ln_g, const float* __restrict__ ln_b,
                const float* __restrict__ wq, const float* __restrict__ bq,
                const float* __restrict__ wo, const float* __restrict__ bo,
                float* __restrict__ out, float* __restrict__ mem_out)
{
    __shared__ float h0s[16 * S0];
    __shared__ float act1[16 * S1];
    __shared__ int   tok[16];
    __shared__ float q[H];
    __shared__ float attn[K_SLOTS];
    __shared__ float pooled[H];

    const int b = blockIdx.x;
    const int lane = threadIdx.x;

    if (lane < 16) {
        int t = 0;
        if (lane < K_SLOTS)        t = seq[(long)b * NL + idx[b * K_SLOTS + lane]];
        else if (lane == K_SLOTS)  t = seq[(long)b * NL + (NL - 1)];
        tok[lane] = t;
    }
    __syncthreads();

    compute_h_tile(tok, embed, W1, b1, W2, b2, ln_g, ln_b, h0s, act1, lane);

    // memory output: rows 0..7 of h
    for (int e = lane; e < K_SLOTS * H; e += 32) {
        int k = e >> 6, c = e & 63;
        mem_out[(long)b * (K_SLOTS * H) + e] = h0s[k * S0 + c];
    }

    // q = h_last @ wq + bq  (row 8 is the last token)
    for (int c = lane; c < H; c += 32) {
        float acc = bq[c];
        for (int k = 0; k < H; ++k) acc += h0s[K_SLOTS * S0 + k] * wq[k * H + c];
        q[c] = acc;
    }
    __syncthreads();

    // logits = memory . q / sqrt(H)
    if (lane < K_SLOTS) {
        float acc = 0.f;
        for (int c = 0; c < H; ++c) acc += h0s[lane * S0 + c] * q[c];
        attn[lane] = acc * 0.125f;           // 1/sqrt(64)
    }
    __syncthreads();

    // softmax over 8 slots
    if (lane == 0) {
        float m = attn[0];
        for (int k = 1; k < K_SLOTS; ++k) m = attn[k] > m ? attn[k] : m;
        float s = 0.f;
        for (int k = 0; k < K_SLOTS; ++k) { float e = __expf(attn[k] - m); attn[k] = e; s += e; }
        float inv = 1.f / s;
        for (int k = 0; k < K_SLOTS; ++k) attn[k] *= inv;
    }
    __syncthreads();

    // pooled = attn @ memory
    for (int c = lane; c < H; c += 32) {
        float acc = 0.f;
        for (int k = 0; k < K_SLOTS; ++k) acc += attn[k] * h0s[k * S0 + c];
        pooled[c] = acc;
    }
    __syncthreads();

    // out = pooled @ wo + bo
    for (int v = lane; v < VOCAB; v += 32) {
        float acc = bo[v];
        for (int hh = 0; hh < H; ++hh) acc += pooled[hh] * wo[hh * VOCAB + v];
        out[(long)b * VOCAB + v] = acc;
    }
}

extern "C" void kernel_launch(void* const* d_in, const int* in_sizes, int n_in,
                              void* d_out, int out_size, void* d_ws, size_t ws_size,
                              hipStream_t stream)
{
    const int*   seq    = (const int*)  d_in[0];
    const float* embed  = (const float*)d_in[1];
    const float* W1     = (const float*)d_in[2];
    const float* b1     = (const float*)d_in[3];
    const float* W2     = (const float*)d_in[4];
    const float* b2     = (const float*)d_in[5];
    const float* ln_g   = (const float*)d_in[6];
    const float* ln_b   = (const float*)d_in[7];
    const float* gate_w = (const float*)d_in[8];
    const float* gate_b = (const float*)d_in[9];
    const float* wq     = (const float*)d_in[10];
    const float* bq     = (const float*)d_in[11];
    const float* wo     = (const float*)d_in[12];
    const float* bo     = (const float*)d_in[13];

    float* out     = (float*)d_out;                 // [128,64]  (first tuple element)
    float* mem_out = (float*)d_out + NB * VOCAB;    // [128,8,64] (second tuple element)

    float* scores = (float*)d_ws;                                          // 2 MB, L2-resident
    int*   idx    = (int*)((char*)d_ws + (size_t)NB * NL * sizeof(float)); // 4 KB

    // 32768 16-token tiles, 4 waves (tiles) per 128-thread block
    score_kernel<<<(NB * NL) / 64, 128, 0, stream>>>(
        seq, embed, W1, b1, W2, b2, ln_g, ln_b, gate_w, gate_b, scores);

    topk_kernel<<<NB, 256, 0, stream>>>(scores, idx);

    finalize_kernel<<<NB, 32, 0, stream>>>(
        seq, idx, embed, W1, b1, W2, b2, ln_g, ln_b, wq, bq, wo, bo, out, mem_out);
}